// MetapathContextEncoder_37452114821486
// MI455X (gfx1250) — compile-verified
//
#include <hip/hip_runtime.h>
#include <hip/hip_bf16.h>

#define D      128
#define H      8
#define DK     16
#define SCALE  0.25f   // 1/sqrt(16)

typedef __attribute__((ext_vector_type(2))) float v2f;
typedef __attribute__((ext_vector_type(8))) float v8f;

// ---------- ordered-uint encoding for float atomic max ----------
__device__ __forceinline__ unsigned fenc(float x) {
  unsigned u = __float_as_uint(x);
  return (u & 0x80000000u) ? ~u : (u | 0x80000000u);
}
__device__ __forceinline__ float fdec(unsigned k) {
  return __uint_as_float((k & 0x80000000u) ? (k & 0x7FFFFFFFu) : ~k);
}

// ---------- Y[M,128] = X[M,128] @ W[128,128]^T  via fp32 WMMA ----------
// One wave computes a 16-row x 128-col block: 8 accumulator tiles, A reused
// across the 8 n-tiles. Software-pipelined: next k-step's A + 8 B fragments
// are loaded while the current 8 WMMAs execute, so waits are partial instead
// of full stalls.
__global__ void gemm_xwT_wmma(const float* __restrict__ X,
                              const float* __restrict__ W,
                              float* __restrict__ Y, int M) {
  const int lane  = threadIdx.x & 31;
  const int wave  = (blockIdx.x * blockDim.x + threadIdx.x) >> 5;
  const int mBase = wave * 16;
  if (mBase >= M) return;

  const int row  = lane & 15;          // M row (A) / N col (B) within tile
  const int koff = (lane >> 4) << 1;   // lanes 0-15 hold K0,K1 ; lanes 16-31 hold K2,K3

  const float* xrow  = X + (size_t)(mBase + row) * D + koff;
  const float* wbase = W + (size_t)row * D + koff;   // + t*16*D selects n-tile row

  v8f acc[8] = {};

  // prologue: fragments for k0 = 0
  v2f a_cur = *(const v2f*)(xrow);
  v2f b_cur[8];
#pragma unroll
  for (int t = 0; t < 8; ++t)
    b_cur[t] = *(const v2f*)(wbase + (size_t)t * 16 * D);

#pragma unroll 4
  for (int k0 = 0; k0 < D; k0 += 4) {
    v2f a_nxt;
    v2f b_nxt[8];
    if (k0 + 4 < D) {
      // issue next-step loads before consuming current fragments
      a_nxt = *(const v2f*)(xrow + k0 + 4);
#pragma unroll
      for (int t = 0; t < 8; ++t)
        b_nxt[t] = *(const v2f*)(wbase + (size_t)t * 16 * D + k0 + 4);
    }
#pragma unroll
    for (int t = 0; t < 8; ++t) {
      // (neg_a, A, neg_b, B, c_mod, C, reuse_a, reuse_b)
      acc[t] = __builtin_amdgcn_wmma_f32_16x16x4_f32(
          false, a_cur, false, b_cur[t], (short)0, acc[t], false, false);
    }
    a_cur = a_nxt;
#pragma unroll
    for (int t = 0; t < 8; ++t) b_cur[t] = b_nxt[t];
  }

  // C/D layout: VGPR r -> lanes 0-15: M=r, N=lane ; lanes 16-31: M=r+8, N=lane-16
  const int n    = lane & 15;
  const int mAdd = (lane >> 4) << 3;
  float* yb = Y + (size_t)mBase * D;
#pragma unroll
  for (int t = 0; t < 8; ++t) {
#pragma unroll
    for (int r = 0; r < 8; ++r) {
      yb[(size_t)(mAdd + r) * D + t * 16 + n] = acc[t][r];
    }
  }
}

// ---------- per-(dst,head): self score, seed running max ----------
__global__ void self_init(const float* __restrict__ dstK,
                          const float* __restrict__ Q,
                          float* __restrict__ self_t,
                          unsigned* __restrict__ m_enc, int Ndst) {
  int tid = blockIdx.x * blockDim.x + threadIdx.x;
  if (tid >= Ndst * H) return;
  int n = tid >> 3, h = tid & 7;
  const float4* kp = (const float4*)(dstK + (size_t)n * D + h * DK);
  const float4* qp = (const float4*)(Q    + (size_t)n * D + h * DK);
  float acc = 0.f;
#pragma unroll
  for (int i = 0; i < 4; ++i) {
    float4 kv = kp[i], qv = qp[i];
    acc += kv.x * qv.x + kv.y * qv.y + kv.z * qv.z + kv.w * qv.w;
  }
  float st = acc * SCALE;
  self_t[tid] = st;
  m_enc[tid]  = fenc(st);
}

// ---------- per-(edge,head): score t, atomic max ----------
__global__ void edge_scores(const float* __restrict__ K,
                            const float* __restrict__ Q,
                            const int* __restrict__ src_idx,
                            const int* __restrict__ dst_idx,
                            float* __restrict__ t_buf,
                            unsigned* __restrict__ m_enc, int E) {
  int tid = blockIdx.x * blockDim.x + threadIdx.x;
  if (tid >= E * H) return;
  int e = tid >> 3, h = tid & 7;
  int s = src_idx[e], d = dst_idx[e];
  const float4* kp = (const float4*)(K + (size_t)s * D + h * DK);
  const float4* qp = (const float4*)(Q + (size_t)d * D + h * DK);
  float acc = 0.f;
#pragma unroll
  for (int i = 0; i < 4; ++i) {
    float4 kv = kp[i], qv = qp[i];
    acc += kv.x * qv.x + kv.y * qv.y + kv.z * qv.z + kv.w * qv.w;
  }
  float t = acc * SCALE;
  t_buf[tid] = t;
  atomicMax(&m_enc[d * H + h], fenc(t));
}

// ---------- per-(dst,head): denom = self_e ----------
__global__ void denom_init(const float* __restrict__ self_t,
                           const unsigned* __restrict__ m_enc,
                           float* __restrict__ denom, int Ndst) {
  int tid = blockIdx.x * blockDim.x + threadIdx.x;
  if (tid >= Ndst * H) return;
  denom[tid] = __expf(self_t[tid] - fdec(m_enc[tid]));
}

// ---------- per-(edge,head): e = exp(t-m), accumulate denom ----------
__global__ void edge_exp(const int* __restrict__ dst_idx,
                         const unsigned* __restrict__ m_enc,
                         float* __restrict__ t_buf,
                         float* __restrict__ denom, int E) {
  int tid = blockIdx.x * blockDim.x + threadIdx.x;
  if (tid >= E * H) return;
  int e = tid >> 3, h = tid & 7;
  int d = dst_idx[e];
  float ev = __expf(t_buf[tid] - fdec(m_enc[d * H + h]));
  t_buf[tid] = ev;
  atomicAdd(&denom[d * H + h], ev);
}

// ---------- per-(dst,head): denom -> 1/denom ----------
__global__ void recip_denom(float* __restrict__ denom, int Ndst) {
  int tid = blockIdx.x * blockDim.x + threadIdx.x;
  if (tid >= Ndst * H) return;
  denom[tid] = 1.0f / denom[tid];
}

// ---------- per-(edge,element): scatter a * h_src into out ----------
__global__ void edge_aggregate(const float* __restrict__ h_src,
                               const int* __restrict__ src_idx,
                               const int* __restrict__ dst_idx,
                               const float* __restrict__ t_buf,
                               const float* __restrict__ rdenom,
                               float* __restrict__ out, int E) {
  unsigned tid = blockIdx.x * blockDim.x + threadIdx.x;
  if (tid >= (unsigned)E * D) return;
  int e = tid >> 7, c = tid & (D - 1), h = c >> 4;
  int s = src_idx[e], d = dst_idx[e];
  float a = t_buf[e * H + h] * rdenom[d * H + h];
  atomicAdd(&out[(size_t)d * D + c], a * h_src[(size_t)s * D + c]);
}

// ---------- per-(dst,element): add self-loop term ----------
__global__ void finalize(const float* __restrict__ h_dst,
                         const float* __restrict__ self_t,
                         const unsigned* __restrict__ m_enc,
                         const float* __restrict__ rdenom,
                         float* __restrict__ out, int Ndst) {
  int tid = blockIdx.x * blockDim.x + threadIdx.x;
  if (tid >= Ndst * D) return;
  int n = tid >> 7, c = tid & (D - 1), h = c >> 4;
  float m = fdec(m_enc[n * H + h]);
  float dst_a = __expf(self_t[n * H + h] - m) * rdenom[n * H + h];
  out[tid] += dst_a * h_dst[tid];
}

extern "C" void kernel_launch(void* const* d_in, const int* in_sizes, int n_in,
                              void* d_out, int out_size, void* d_ws, size_t ws_size,
                              hipStream_t stream) {
  const float* h_src = (const float*)d_in[0];
  const float* h_dst = (const float*)d_in[1];
  const float* Wk    = (const float*)d_in[2];
  const float* Wq    = (const float*)d_in[3];
  const int* src_idx = (const int*)d_in[4];
  const int* dst_idx = (const int*)d_in[5];

  const int N_SRC = in_sizes[0] / D;   // 200000
  const int N_DST = in_sizes[1] / D;   // 50000
  const int E     = in_sizes[4];       // 1600000

  // workspace carve-up (256B aligned)
  auto align256 = [](size_t x) { return (x + 255) & ~(size_t)255; };
  char* ws = (char*)d_ws;
  size_t off = 0;
  float* K      = (float*)(ws + off); off += align256((size_t)N_SRC * D * 4);
  float* Q      = (float*)(ws + off); off += align256((size_t)N_DST * D * 4);
  float* dstK   = (float*)(ws + off); off += align256((size_t)N_DST * D * 4);
  float* t_buf  = (float*)(ws + off); off += align256((size_t)E * H * 4);
  float* self_t = (float*)(ws + off); off += align256((size_t)N_DST * H * 4);
  unsigned* m_enc = (unsigned*)(ws + off); off += align256((size_t)N_DST * H * 4);
  float* denom  = (float*)(ws + off); off += align256((size_t)N_DST * H * 4);

  const int TB = 256;

  hipMemsetAsync(d_out, 0, (size_t)out_size * sizeof(float), stream);

  // projections via fp32 WMMA (wave = 16 rows x 128 cols)
  {
    int waves  = N_SRC / 16;
    int blocks = (waves * 32 + TB - 1) / TB;
    gemm_xwT_wmma<<<blocks, TB, 0, stream>>>(h_src, Wk, K, N_SRC);
  }
  {
    int waves  = N_DST / 16;
    int blocks = (waves * 32 + TB - 1) / TB;
    gemm_xwT_wmma<<<blocks, TB, 0, stream>>>(h_dst, Wq, Q, N_DST);
    gemm_xwT_wmma<<<blocks, TB, 0, stream>>>(h_dst, Wk, dstK, N_DST);
  }

  self_init<<<(N_DST * H + TB - 1) / TB, TB, 0, stream>>>(dstK, Q, self_t, m_enc, N_DST);
  edge_scores<<<(E * H + TB - 1) / TB, TB, 0, stream>>>(K, Q, src_idx, dst_idx, t_buf, m_enc, E);
  denom_init<<<(N_DST * H + TB - 1) / TB, TB, 0, stream>>>(self_t, m_enc, denom, N_DST);
  edge_exp<<<(E * H + TB - 1) / TB, TB, 0, stream>>>(dst_idx, m_enc, t_buf, denom, E);
  recip_denom<<<(N_DST * H + TB - 1) / TB, TB, 0, stream>>>(denom, N_DST);
  {
    unsigned total  = (unsigned)E * D;
    unsigned blocks = (total + TB - 1) / TB;
    edge_aggregate<<<blocks, TB, 0, stream>>>(h_src, src_idx, dst_idx, t_buf, denom, (float*)d_out, E);
  }
  finalize<<<(N_DST * D + TB - 1) / TB, TB, 0, stream>>>(h_dst, self_t, m_enc, denom, (float*)d_out, N_DST);
}